// CenterLoss_7370163880488
// MI455X (gfx1250) — compile-verified
//
#include <hip/hip_runtime.h>

// CenterLoss on MI455X (gfx1250, wave32).
// loss = (1/B) * sum_b sum_d (features[b,d] - centers[labels[b],d])^2,  LAMBDA_C = 1
//
// Memory-bound: ~128-166 MB @ 23.3 TB/s ~= 6 us. Strategy:
//  - wave32 per row, float4 coalesced loads (lane l -> bytes 16l..16l+15)
//  - row / label forced wave-uniform via readfirstlane so the label fetch
//    scalarizes to SMEM (KMcnt) and never drains the vector-load pipe (LOADcnt)
//  - features loaded non-temporal (64 MB stream) so the 102.4 MB centers table
//    stays resident in the 192 MB L2 across the random gather
//  - next-label software pipeline + global_prefetch of next center row (GL2)
//  - cross-lane fold via V_WMMA_F32_16X16X4_F32 with B = ones (exact f32)
//  - deterministic 2-stage reduction through d_ws (no float atomics)

#define BATCH        65536
#define FEAT         256
#define BLOCK        256
#define NBLOCKS      1024
#define WAVES_PER_BLOCK (BLOCK / 32)
#define TOTAL_WAVES  (NBLOCKS * WAVES_PER_BLOCK)   // 8192 waves -> 8 rows/wave

typedef __attribute__((ext_vector_type(2))) float v2f;
typedef __attribute__((ext_vector_type(4))) float v4f;
typedef __attribute__((ext_vector_type(8))) float v8f;

__global__ __launch_bounds__(BLOCK)
void center_loss_partial(const float* __restrict__ features,
                         const int*   __restrict__ labels,
                         const float* __restrict__ centers,
                         float*       __restrict__ partials)
{
    const int lane = threadIdx.x & 31;
    const int wid  = threadIdx.x >> 5;
    // Wave-uniform row cursor: readfirstlane lets the compiler scalarize all
    // row/label address math (SALU + SMEM instead of VALU + VMEM).
    const int gwave = __builtin_amdgcn_readfirstlane(
                          (int)(blockIdx.x * WAVES_PER_BLOCK + wid));

    float acc = 0.0f;

    int row = gwave;
    int next_label = (row < BATCH)
                   ? __builtin_amdgcn_readfirstlane(labels[row]) : 0;

    for (; row < BATCH; row += TOTAL_WAVES) {
        const int label = next_label;
        const int nrow  = row + TOTAL_WAVES;
        if (nrow < BATCH) {
            // Scalar (SMEM/KMcnt) label fetch for the next row + GL2 prefetch
            // of its center row; does not stall the LOADcnt stream below.
            next_label = __builtin_amdgcn_readfirstlane(labels[nrow]);
            __builtin_prefetch(centers + (size_t)next_label * FEAT, 0, 2);
        }

        const v4f* __restrict__ f = (const v4f*)(features + (size_t)row   * FEAT);
        const v4f* __restrict__ c = (const v4f*)(centers  + (size_t)label * FEAT);

        // 256 floats/row, 32 lanes -> 2x float4 per lane, fully coalesced
        // (uniform SGPR base + per-lane offset).
        v4f a0 = __builtin_nontemporal_load(f + lane);        // streaming: NT hint
        v4f a1 = __builtin_nontemporal_load(f + lane + 32);
        v4f b0 = c[lane];                                     // L2-resident gather
        v4f b1 = c[lane + 32];

        v4f d0 = a0 - b0;
        v4f d1 = a1 - b1;
        acc = fmaf(d0.x, d0.x, acc);
        acc = fmaf(d0.y, d0.y, acc);
        acc = fmaf(d0.z, d0.z, acc);
        acc = fmaf(d0.w, d0.w, acc);
        acc = fmaf(d1.x, d1.x, acc);
        acc = fmaf(d1.y, d1.y, acc);
        acc = fmaf(d1.z, d1.z, acc);
        acc = fmaf(d1.w, d1.w, acc);
    }

    // ---- Cross-lane fold with V_WMMA_F32_16X16X4_F32, B = ones ----
    // A layout (32-bit 16x4): lane m (<16): vgpr0=K0, vgpr1=K1; lane m+16: K2,K3.
    // With A = {acc, 0} per lane and B = 1 everywhere:
    //   D[m,n] = acc[m] + acc[m+16]  (replicated over n), exact f32.
    v2f A; A[0] = acc;  A[1] = 0.0f;
    v2f Bm; Bm[0] = 1.0f; Bm[1] = 1.0f;
    v8f C = {};
    C = __builtin_amdgcn_wmma_f32_16x16x4_f32(
            /*neg_a=*/false, A, /*neg_b=*/false, Bm,
            /*c_mod=*/(short)0, C, /*reuse_a=*/false, /*reuse_b=*/false);

    // D vgpr r holds rows r (lanes 0-15) and 8+r (lanes 16-31): sum the 8 vgprs,
    // then fold the two half-wave sums with one xor-16 shuffle.
    float t = C[0] + C[1] + C[2] + C[3] + C[4] + C[5] + C[6] + C[7];
    t += __shfl_xor(t, 16, 32);

    __shared__ float smem[WAVES_PER_BLOCK];
    if (lane == 0) smem[wid] = t;
    __syncthreads();
    if (threadIdx.x == 0) {
        float s = 0.0f;
#pragma unroll
        for (int i = 0; i < WAVES_PER_BLOCK; ++i) s += smem[i];
        partials[blockIdx.x] = s;
    }
}

__global__ __launch_bounds__(BLOCK)
void center_loss_finalize(const float* __restrict__ partials,
                          float*       __restrict__ out)
{
    __shared__ float smem[BLOCK];
    float s = 0.0f;
    for (int i = threadIdx.x; i < NBLOCKS; i += BLOCK) s += partials[i];
    smem[threadIdx.x] = s;
    __syncthreads();
#pragma unroll
    for (int off = BLOCK / 2; off > 0; off >>= 1) {
        if (threadIdx.x < off) smem[threadIdx.x] += smem[threadIdx.x + off];
        __syncthreads();
    }
    if (threadIdx.x == 0)
        out[0] = smem[0] * (1.0f / (float)BATCH);   // LAMBDA_C == 1.0
}

extern "C" void kernel_launch(void* const* d_in, const int* in_sizes, int n_in,
                              void* d_out, int out_size, void* d_ws, size_t ws_size,
                              hipStream_t stream)
{
    const float* features = (const float*)d_in[0];   // [65536, 256] f32
    const int*   labels   = (const int*)  d_in[1];   // [65536]      i32
    const float* centers  = (const float*)d_in[2];   // [100000, 256] f32
    float*       out      = (float*)d_out;           // scalar f32
    float*       partials = (float*)d_ws;            // NBLOCKS floats (4 KB)

    center_loss_partial<<<NBLOCKS, BLOCK, 0, stream>>>(features, labels, centers, partials);
    center_loss_finalize<<<1, BLOCK, 0, stream>>>(partials, out);
}